// TPS_283467842274
// MI455X (gfx1250) — compile-verified
//
#include <hip/hip_runtime.h>
#include <math.h>

#define HGT 512
#define WID 512
#define NPTS 64
#define LSZ 67            // 64 + 3
#define NCOL 69           // 67 cols + 2 RHS
#define EPSV 1e-9f
#define TILES_PER_BATCH (HGT * WID / 16)        // 16384
#define BLOCKS_PER_BATCH (TILES_PER_BATCH / 8)  // 2048

typedef __attribute__((ext_vector_type(2))) float v2f;
typedef __attribute__((ext_vector_type(4))) float v4f;
typedef __attribute__((ext_vector_type(8))) float v8f;

// Hardware v_log_f32 (log2, ~1 ulp) scaled by ln2. Valid: args are >= EPS,
// far above the denormal threshold, so no fixup path needed.
__device__ __forceinline__ float fast_logf(float x) {
    return __builtin_amdgcn_logf(x) * 0.69314718055994531f;
}

// ---------------------------------------------------------------------------
// Kernel 1: build L = [[K, P],[P^T, 0]], Z = [Y;0]; solve L Q = Z per batch
// via Gaussian elimination with partial pivoting (K diagonal is zero, so
// pivoting is required). One workgroup per batch; whole system lives in LDS.
// ---------------------------------------------------------------------------
__global__ __launch_bounds__(128) void tps_solve_kernel(
    const float* __restrict__ X,   // [B][64][2]
    const float* __restrict__ Y,   // [B][64][2]
    float* __restrict__ Q)         // [B][67][2]  (workspace)
{
    __shared__ float M[LSZ][NCOL + 1];   // augmented [L | Z], padded row
    __shared__ float sx[NPTS][2];
    __shared__ float fac[LSZ];
    __shared__ float Qs[LSZ][2];
    __shared__ int piv;

    const int b   = blockIdx.x;
    const int tid = threadIdx.x;
    const int nt  = blockDim.x;

    for (int i = tid; i < NPTS * 2; i += nt)
        sx[i >> 1][i & 1] = X[b * NPTS * 2 + i];
    __syncthreads();

    // Build augmented matrix
    for (int idx = tid; idx < LSZ * NCOL; idx += nt) {
        const int r = idx / NCOL, c = idx % NCOL;
        float v = 0.0f;
        if (c < LSZ) {
            if (r < NPTS && c < NPTS) {
                const float dx = sx[r][0] - sx[c][0];
                const float dy = sx[r][1] - sx[c][1];
                const float d2 = dx * dx + dy * dy;
                v = d2 * fast_logf(d2 + EPSV);
            } else if (r < NPTS) {              // P block
                v = (c == NPTS) ? 1.0f : sx[r][c - NPTS - 1];
            } else if (c < NPTS) {              // P^T block
                v = (r == NPTS) ? 1.0f : sx[c][r - NPTS - 1];
            }                                    // else 3x3 zero block
        } else {
            const int n = c - LSZ;
            v = (r < NPTS) ? Y[(b * NPTS + r) * 2 + n] : 0.0f;
        }
        M[r][c] = v;
    }
    __syncthreads();

    // Forward elimination with partial pivoting
    for (int col = 0; col < LSZ; ++col) {
        if (tid == 0) {
            int p = col;
            float best = fabsf(M[col][col]);
            for (int r = col + 1; r < LSZ; ++r) {
                const float a = fabsf(M[r][col]);
                if (a > best) { best = a; p = r; }
            }
            piv = p;
        }
        __syncthreads();
        const int p = piv;
        if (p != col) {
            for (int j = tid; j < NCOL; j += nt) {
                const float t = M[col][j]; M[col][j] = M[p][j]; M[p][j] = t;
            }
        }
        __syncthreads();
        const float pivval = M[col][col];
        for (int r = col + 1 + tid; r < LSZ; r += nt)
            fac[r] = M[r][col] / pivval;
        __syncthreads();
        const int nbelow = LSZ - 1 - col;
        const int width  = NCOL - col;
        for (int idx = tid; idx < nbelow * width; idx += nt) {
            const int r = col + 1 + idx / width;
            const int j = col + idx % width;
            M[r][j] -= fac[r] * M[col][j];
        }
        __syncthreads();
    }

    // Back substitution (2 RHS columns, one thread each — trivial cost)
    if (tid < 2) {
        const int n = tid;
        for (int r = LSZ - 1; r >= 0; --r) {
            float s = M[r][LSZ + n];
            for (int j = r + 1; j < LSZ; ++j) s -= M[r][j] * Qs[j][n];
            Qs[r][n] = s / M[r][r];
        }
    }
    __syncthreads();
    for (int i = tid; i < LSZ * 2; i += nt)
        Q[b * LSZ * 2 + i] = Qs[i >> 1][i & 1];
}

// ---------------------------------------------------------------------------
// Kernel 2: fused U-generation + U@Wc via V_WMMA_F32_16X16X4_F32.
// Each wave32 owns a 16-grid-point tile. Per WMMA step (K=4):
//   A (16x4 f32, 2 VGPRs): lane group h (lane>>4), VGPR v hold K = 2h+v,
//     row M = lane&15  -> each lane computes 2 RBF values (2 v_log_f32).
//   B (4x16 f32, 2 VGPRs): pre-swizzled in LDS per (step, lane) so the loop
//     does one unpredicated ds_load_b64 (cols >= 2 zero-padded Wc).
// X rows kb,kb+1 are contiguous -> one ds_load_b128 per step.
// 16 steps accumulate K=64 into the f32 C tile; affine Pg@A added at store.
// ---------------------------------------------------------------------------
__global__ __launch_bounds__(256) void tps_eval_kernel(
    const float* __restrict__ X,   // [B][64][2]
    const float* __restrict__ Q,   // [B][67][2]
    float* __restrict__ out)       // [B][512*512][2]
{
    __shared__ float sX[NPTS * 2];                       // [64][2] row-major
    __shared__ float sWB[16][32][2];                     // B operand, WMMA layout
    __shared__ float sA[3][2];

    const int b     = blockIdx.x / BLOCKS_PER_BATCH;
    const int tile0 = (blockIdx.x % BLOCKS_PER_BATCH) * 8;
    const int wave  = threadIdx.x >> 5;
    const int lane  = threadIdx.x & 31;
    const int h     = lane >> 4;      // half-wave group
    const int nm    = lane & 15;      // A-row index == B/C column index

    for (int i = threadIdx.x; i < NPTS * 2; i += blockDim.x)
        sX[i] = X[b * NPTS * 2 + i];
    // B operand swizzle: step s, lane l, vgpr t  <-  Wc[s*4 + 2*(l>>4) + t][l&15]
    for (int i = threadIdx.x; i < 16 * 32 * 2; i += blockDim.x) {
        const int s = i >> 6, l = (i >> 1) & 31, t = i & 1;
        const int ln = l & 15, lh = l >> 4;
        const int k  = s * 4 + 2 * lh + t;
        sWB[s][l][t] = (ln < 2) ? Q[b * LSZ * 2 + k * 2 + ln] : 0.0f;
    }
    if (threadIdx.x < 6)
        sA[threadIdx.x >> 1][threadIdx.x & 1] = Q[b * LSZ * 2 + NPTS * 2 + threadIdx.x];
    __syncthreads();

    const int   base = (tile0 + wave) * 16;
    const float step = 2.0f / 511.0f;
    const int   pm   = base + nm;                    // this lane's A-matrix row
    const float gxm  = -1.0f + (float)(pm & 511) * step;
    const float gym  = -1.0f + (float)(pm >> 9) * step;

    v8f c = {0.f, 0.f, 0.f, 0.f, 0.f, 0.f, 0.f, 0.f};
#pragma unroll
    for (int s = 0; s < 16; ++s) {
        // X rows kb = s*4+2h and kb+1: 4 contiguous, 16B-aligned floats
        const v4f xv = *(const v4f*)(&sX[8 * s + 4 * h]);
        const float dx0 = gxm - xv.x, dy0 = gym - xv.y;
        const float dx1 = gxm - xv.z, dy1 = gym - xv.w;
        const float d20 = dx0 * dx0 + dy0 * dy0;
        const float d21 = dx1 * dx1 + dy1 * dy1;
        v2f a;
        a[0] = d20 * fast_logf(d20 + EPSV);
        a[1] = d21 * fast_logf(d21 + EPSV);
        const v2f bv = *(const v2f*)(&sWB[s][lane][0]);
        // D = A(16x4) * B(4x16) + C, full f32
        c = __builtin_amdgcn_wmma_f32_16x16x4_f32(
                /*neg_a=*/false, a, /*neg_b=*/false, bv,
                /*c_mod=*/(short)0, c, /*reuse_a=*/false, /*reuse_b=*/false);
    }

    // Store valid columns (n = 0,1), add affine term Pg@A
    if (nm < 2) {
        const float a0 = sA[0][nm], a1 = sA[1][nm], a2 = sA[2][nm];
#pragma unroll
        for (int v = 0; v < 8; ++v) {
            const int   p  = base + v + 8 * h;       // C/D row M = v + 8h
            const float gx = -1.0f + (float)(p & 511) * step;
            const float gy = -1.0f + (float)(p >> 9) * step;
            out[(size_t)(b * (HGT * WID) + p) * 2 + nm] =
                c[v] + a0 + gx * a1 + gy * a2;
        }
    }
}

// ---------------------------------------------------------------------------
extern "C" void kernel_launch(void* const* d_in, const int* in_sizes, int n_in,
                              void* d_out, int out_size, void* d_ws, size_t ws_size,
                              hipStream_t stream) {
    (void)n_in; (void)out_size; (void)ws_size;
    const float* X = (const float*)d_in[0];
    const float* Y = (const float*)d_in[1];
    float* out = (float*)d_out;
    float* Q   = (float*)d_ws;                // [B][67][2] scratch

    const int B = in_sizes[0] / (NPTS * 2);   // 8 batches

    tps_solve_kernel<<<B, 128, 0, stream>>>(X, Y, Q);
    tps_eval_kernel<<<B * BLOCKS_PER_BATCH, 256, 0, stream>>>(X, Q, out);
}